// Mvdr_77378130804759
// MI455X (gfx1250) — compile-verified
//
#include <hip/hip_runtime.h>
#include <stdint.h>

#define BINS_N   262144
#define EPS      1e-6f

// ---- per-wave LDS layout (one wave per block, 32 bins per wave) ----
// each staged 8x8 complex64 matrix = 512B; pad to 528B (16B aligned for async
// b128 writes; 132-dword stride spreads LDS banks).
#define BIN_STRIDE 528u
#define N_OFF      0u
#define T_OFF      (32u * BIN_STRIDE)                    // 16896
#define SCR_OFF    (2u * 32u * BIN_STRIDE)               // 33792
#define LDS_BYTES  (SCR_OFF + 32u * 16u * 4u)            // 35840

typedef float v2f __attribute__((ext_vector_type(2)));
typedef float v8f __attribute__((ext_vector_type(8)));

__device__ __forceinline__ constexpr int sidx(int i, int j) { return (i*(i-1))/2 + j; } // strict lower (i>j)
__device__ __forceinline__ constexpr int tidx(int i, int j) { return (i*(i+1))/2 + j; } // lower incl diag (j<=i)

__device__ __forceinline__ float2 cmul(float2 a, float2 b) {            // a*b
  return make_float2(fmaf(a.x,b.x,-a.y*b.y), fmaf(a.x,b.y, a.y*b.x));
}
__device__ __forceinline__ float2 cmadd(float2 s, float2 a, float2 b) { // s + a*b
  return make_float2(fmaf(a.x,b.x, fmaf(-a.y,b.y,s.x)),
                     fmaf(a.x,b.y, fmaf( a.y,b.x,s.y)));
}
__device__ __forceinline__ float2 cmaddc(float2 s, float2 a, float2 b) { // s + a*conj(b)
  return make_float2(fmaf(a.x,b.x, fmaf( a.y,b.y,s.x)),
                     fmaf(a.y,b.x, fmaf(-a.x,b.y,s.y)));
}
__device__ __forceinline__ float2 cmaddca(float2 s, float2 a, float2 b) { // s + conj(a)*b
  return make_float2(fmaf(a.x,b.x, fmaf( a.y,b.y,s.x)),
                     fmaf(a.x,b.y, fmaf(-a.y,b.x,s.y)));
}
__device__ __forceinline__ float2 cnmaddc(float2 s, float2 a, float2 b) { // s - a*conj(b)
  return make_float2(fmaf(-a.x,b.x, fmaf(-a.y,b.y,s.x)),
                     fmaf(-a.y,b.x, fmaf( a.x,b.y,s.y)));
}

// Element (r,c) of the real 16x16 block embedding [[Re,-Im],[Im,Re]] of a
// complex 8x8 matrix stored row-major interleaved (re,im) at M.
__device__ __forceinline__ float bval(const float* M, int r, int c) {
  const int rr = r & 7, cc = c & 7;
  float v = M[((rr << 3) + cc) * 2 + (((r >> 3) ^ (c >> 3)) & 1)];
  return (r < 8 && c >= 8) ? -v : v;   // top-right quadrant is -Im
}

__global__ __launch_bounds__(32) void mvdr_cdna5_kernel(
    const float2* __restrict__ signal,   // (8, BINS)
    const float2* __restrict__ target,   // (BINS, 8, 8)
    const float2* __restrict__ noise,    // (BINS, 8, 8)
    float2* __restrict__ out)            // (1, BINS)
{
  extern __shared__ char smem[];
  const int lane    = threadIdx.x & 31;
  const int binBase = blockIdx.x << 5;
  const unsigned ldsBase = (unsigned)(uintptr_t)smem;  // low 32 bits of generic == LDS offset

  // ---------------- Stage 32 bins of noise/target SCMs into LDS (async) ----
  {
    unsigned long long gN = (unsigned long long)(uintptr_t)noise
                          + (unsigned long long)binBase * 512ull + (unsigned)lane * 16u;
    unsigned long long gT = (unsigned long long)(uintptr_t)target
                          + (unsigned long long)binBase * 512ull + (unsigned)lane * 16u;
    const unsigned lN = ldsBase + N_OFF + (unsigned)lane * 16u;
    const unsigned lT = ldsBase + T_OFF + (unsigned)lane * 16u;
#pragma unroll
    for (int i = 0; i < 32; ++i) {       // iteration i moves bin i (32 lanes x 16B = 512B)
      asm volatile("global_load_async_to_lds_b128 %0, %1, off"
                   :: "v"(lN + (unsigned)i * BIN_STRIDE),
                      "v"(gN + (unsigned long long)i * 512ull) : "memory");
      asm volatile("global_load_async_to_lds_b128 %0, %1, off"
                   :: "v"(lT + (unsigned)i * BIN_STRIDE),
                      "v"(gT + (unsigned long long)i * 512ull) : "memory");
    }
    asm volatile("s_wait_asynccnt 0" ::: "memory");
  }

  const float2* Np = (const float2*)(smem + N_OFF + (unsigned)lane * BIN_STRIDE);
  const float2* Tp = (const float2*)(smem + T_OFF + (unsigned)lane * BIN_STRIDE);

  // ---------------- Per-lane: Cholesky of (N + eps I) -----------------------
  float2 L[28]; float invd[8]; float dia[8];
#pragma unroll
  for (int i = 0; i < 8; ++i) {
    dia[i] = Np[i*8 + i].x + EPS;
#pragma unroll
    for (int j = 0; j < i; ++j) L[sidx(i,j)] = Np[i*8 + j];
  }
#pragma unroll
  for (int j = 0; j < 8; ++j) {
    float v = dia[j];
#pragma unroll
    for (int k = 0; k < j; ++k) { float2 l = L[sidx(j,k)]; v = fmaf(-l.x,l.x, fmaf(-l.y,l.y, v)); }
    const float rs = __builtin_amdgcn_rsqf(v);   // 1/L[j][j]; sqrt itself never needed
    invd[j] = rs;
#pragma unroll
    for (int i = j+1; i < 8; ++i) {
      float2 s = L[sidx(i,j)];
#pragma unroll
      for (int k = 0; k < j; ++k) s = cnmaddc(s, L[sidx(i,k)], L[sidx(j,k)]);
      L[sidx(i,j)] = make_float2(s.x * rs, s.y * rs);
    }
  }

  // ---------------- G = L^{-1} (strictly lower; diag = invd) ----------------
  float2 G[28];
#pragma unroll
  for (int j = 0; j < 8; ++j) {
#pragma unroll
    for (int i = j+1; i < 8; ++i) {
      float2 s = make_float2(L[sidx(i,j)].x * invd[j], L[sidx(i,j)].y * invd[j]);
#pragma unroll
      for (int k = j+1; k < i; ++k) s = cmadd(s, L[sidx(i,k)], G[sidx(k,j)]);
      G[sidx(i,j)] = make_float2(-s.x * invd[i], -s.y * invd[i]);
    }
  }

  // ---------------- H = N^{-1} = G^H G (Hermitian lower) --------------------
  float2 H[36];
#pragma unroll
  for (int i = 0; i < 8; ++i) {
#pragma unroll
    for (int j = 0; j <= i; ++j) {
      float2 s = make_float2(0.f, 0.f);
#pragma unroll
      for (int k = i; k < 8; ++k) {
        const float2 gki = (k == i) ? make_float2(invd[i], 0.f) : G[sidx(k,i)];
        const float2 gkj = (k == j) ? make_float2(invd[j], 0.f) : G[sidx(k,j)];
        s = cmaddca(s, gki, gkj);            // s += conj(G[k][i]) * G[k][j]
      }
      H[tidx(i,j)] = s;
    }
  }

  // ---------------- trace(N^{-1} T) = sum_ij H_ij * T_ji (per-lane) ---------
  float2 tr = make_float2(0.f, 0.f);
#pragma unroll
  for (int i = 0; i < 8; ++i) tr = cmadd(tr, H[tidx(i,i)], Tp[i*8 + i]);
#pragma unroll
  for (int i = 0; i < 8; ++i)
#pragma unroll
    for (int j = 0; j < i; ++j) {
      const float2 hij = H[tidx(i,j)];
      tr = cmadd  (tr, hij, Tp[j*8 + i]);    // H_ij * T_ji
      tr = cmaddca(tr, hij, Tp[i*8 + j]);    // conj(H_ij) * T_ij  (= H_ji * T_ij)
    }

  // ---------------- publish full H to LDS (reuse dead noise region) ---------
  {
    float2* Hw = (float2*)(smem + N_OFF + (unsigned)lane * BIN_STRIDE);
#pragma unroll
    for (int i = 0; i < 8; ++i)
#pragma unroll
      for (int j = 0; j < 8; ++j) {
        const float2 h = (j <= i) ? H[tidx(i,j)]
                                  : make_float2(H[tidx(j,i)].x, -H[tidx(j,i)].y);
        Hw[i*8 + j] = h;
      }
  }

  // ---------------- WMMA: P_hat = H_hat x T_hat, one 16x16x16 per bin -------
  // A 16x4 f32 layout: lanes 0-15 supply K=4s+{0,1}, lanes 16-31 K=4s+{2,3};
  // B mirrors the K-halving. Column 0 of P (= beamformer numerator) lives in
  // lanes N=0 (lane 0: Re rows 0-7, lane 16: Im rows 8-15).
  const int m     = lane & 15;
  const int halfk = (lane >> 4) << 1;       // 0 or 2
  float* scrw = (float*)(smem + SCR_OFF);
#pragma unroll
  for (int b = 0; b < 32; ++b) {
    const float* Hb = (const float*)(smem + N_OFF + (unsigned)b * BIN_STRIDE);
    const float* Tb = (const float*)(smem + T_OFF + (unsigned)b * BIN_STRIDE);
    v8f acc = {0.f,0.f,0.f,0.f,0.f,0.f,0.f,0.f};
#pragma unroll
    for (int s4 = 0; s4 < 4; ++s4) {
      const int ka = s4*4 + halfk;
      v2f av, bv;
      av.x = bval(Hb, m, ka);   av.y = bval(Hb, m, ka + 1);
      bv.x = bval(Tb, ka, m);   bv.y = bval(Tb, ka + 1, m);
      acc = __builtin_amdgcn_wmma_f32_16x16x4_f32(false, av, false, bv,
                                                  (short)0, acc, false, false);
    }
    if ((lane & 15) == 0) {                 // lanes 0 and 16 hold column N=0
      float* dp = scrw + b*16 + (lane >> 1);  // lane0 -> +0 (Re), lane16 -> +8 (Im)
#pragma unroll
      for (int r = 0; r < 8; ++r) dp[r] = acc[r];
    }
  }

  // ---------------- epilogue: normalize, phase, apply beamformer ------------
  const float* sp = scrw + lane * 16;
  float2 w[8];
  {
    const float2 denom = make_float2(tr.x + EPS, tr.y);
    const float  q     = __builtin_amdgcn_rcpf(fmaf(denom.x, denom.x, denom.y * denom.y));
    const float2 dinv  = make_float2(denom.x * q, -denom.y * q);
#pragma unroll
    for (int c = 0; c < 8; ++c) {
      const float2 pc = make_float2(sp[c], sp[8 + c]);   // (N^{-1}T)[c][0]
      w[c] = cmul(pc, dinv);
    }
  }
  const float mag  = sqrtf(fmaf(w[0].x, w[0].x, w[0].y * w[0].y));
  const float isc  = __builtin_amdgcn_rcpf(mag + EPS);
  const float2 ph  = make_float2(w[0].x * isc, -w[0].y * isc);  // conj(w0/(|w0|+eps))

  const int f = binBase + lane;
  float2 o = make_float2(0.f, 0.f);
#pragma unroll
  for (int c = 0; c < 8; ++c) {
    const float2 wc = cmul(w[c], ph);
    const float2 y  = signal[(size_t)c * BINS_N + f];
    o = cmaddc(o, y, wc);                   // o += y * conj(wc)
  }
  out[f] = o;
}

extern "C" void kernel_launch(void* const* d_in, const int* in_sizes, int n_in,
                              void* d_out, int out_size, void* d_ws, size_t ws_size,
                              hipStream_t stream) {
  (void)in_sizes; (void)n_in; (void)out_size; (void)d_ws; (void)ws_size;
  const float2* signal = (const float2*)d_in[0];
  const float2* target = (const float2*)d_in[1];
  const float2* noise  = (const float2*)d_in[2];
  float2* o = (float2*)d_out;
  mvdr_cdna5_kernel<<<dim3(BINS_N / 32), dim3(32), LDS_BYTES, stream>>>(signal, target, noise, o);
}